// GraphNormv2_87754771792653
// MI455X (gfx1250) — compile-verified
//
#include <hip/hip_runtime.h>

typedef __attribute__((ext_vector_type(2))) float v2f;
typedef __attribute__((ext_vector_type(8))) float v8f;

// D = A(16x4 f32) * B(4x16 f32) + C(16x16 f32)
#define WMMA_F32(a, b, c) \
    __builtin_amdgcn_wmma_f32_16x16x4_f32(false, (a), false, (b), (short)0, (c), false, false)

constexpr int   N_NODES = 262144;
constexpr int   H       = 256;
constexpr int   E       = 32;
constexpr int   G       = 1024;
constexpr float EPS     = 1e-5f;

// ---------------------------------------------------------------- K0: zero ws
__global__ void zero_kernel(float* __restrict__ p, int n) {
    int i = blockIdx.x * blockDim.x + threadIdx.x;
    if (i < n) p[i] = 0.0f;
}

// ------------------------------------------- K2: contribute[E,H] = ev^T @ x
// Grid: 256 blocks x 256 threads (8 waves). Each block owns 1024 nodes of K.
// Wave w owns H columns [32w, 32w+32). M (=E) split into two 16-tiles.
constexpr int K2_BLOCKS = 256;
constexpr int K2_CHUNK  = N_NODES / K2_BLOCKS;   // 1024 nodes

__global__ void __launch_bounds__(256)
contribute_kernel(const float* __restrict__ x, const float* __restrict__ ev,
                  float* __restrict__ contrib) {
    const int lane = threadIdx.x & 31;
    const int wave = threadIdx.x >> 5;
    const int hb   = wave * 32;
    const int lm   = lane & 15;
    const int hi16 = lane >> 4;        // 0 for lanes 0-15, 1 for lanes 16-31
    const int koff = hi16 << 1;        // K offset 0 or 2 (A/B f32 layout)
    const int kchunk0 = blockIdx.x * K2_CHUNK;

    v8f c00{}, c01{}, c10{}, c11{};
    for (int kk = 0; kk < K2_CHUNK; kk += 4) {
        const int k0 = kchunk0 + kk + koff;
        // A = ev^T : A[m=e][k=node]; lane holds (m=lm, k=koff+{0,1})
        const float* evr0 = ev + (size_t)k0 * E;
        const float* evr1 = evr0 + E;
        v2f a0 = { evr0[lm],      evr1[lm]      };   // e-tile 0
        v2f a1 = { evr0[lm + 16], evr1[lm + 16] };   // e-tile 1
        // B = x : B[k=node][n=h]; lane holds (n=lm, k=koff+{0,1})
        const float* xr0 = x + (size_t)k0 * H + hb;
        const float* xr1 = xr0 + H;
        v2f b0 = { xr0[lm],      xr1[lm]      };     // h-tile 0
        v2f b1 = { xr0[lm + 16], xr1[lm + 16] };     // h-tile 1
        c00 = WMMA_F32(a0, b0, c00);
        c01 = WMMA_F32(a0, b1, c01);
        c10 = WMMA_F32(a1, b0, c10);
        c11 = WMMA_F32(a1, b1, c11);
    }
    // C layout: vgpr d, lane l -> row d + 8*(l>=16), col l%16
    #pragma unroll
    for (int d = 0; d < 8; ++d) {
        const int er = d + (hi16 << 3);
        const int hc = hb + lm;
        atomicAdd(&contrib[ er       * H + hc     ], c00[d]);
        atomicAdd(&contrib[ er       * H + hc + 16], c01[d]);
        atomicAdd(&contrib[(er + 16) * H + hc     ], c10[d]);
        atomicAdd(&contrib[(er + 16) * H + hc + 16], c11[d]);
    }
}

// ------------------------------- K3: scaled = (1 + ev_scales) * contribute
__global__ void scale_kernel(const float* __restrict__ contrib,
                             const float* __restrict__ ev_scales,
                             float* __restrict__ scaled, int n) {
    int i = blockIdx.x * blockDim.x + threadIdx.x;
    if (i < n) scaled[i] = (1.0f + ev_scales[i]) * contrib[i];
}

// ----------------- K4: per-graph fused mean/variance/normalize (1 block/graph)
__global__ void __launch_bounds__(256)
norm_kernel(const float* __restrict__ x, const float* __restrict__ ev,
            const int* __restrict__ batch, const float* __restrict__ weight,
            const float* __restrict__ bias, const float* __restrict__ scaled,
            float* __restrict__ out) {
    __shared__ int s_bounds[2];
    const int g = blockIdx.x;
    if (threadIdx.x == 0) {
        int lo = 0, hi = N_NODES;                        // lower_bound(batch, g)
        while (lo < hi) { int m = (lo + hi) >> 1; if (batch[m] < g) lo = m + 1; else hi = m; }
        s_bounds[0] = lo;
        int lo2 = lo, hi2 = N_NODES;                     // lower_bound(batch, g+1)
        while (lo2 < hi2) { int m = (lo2 + hi2) >> 1; if (batch[m] < g + 1) lo2 = m + 1; else hi2 = m; }
        s_bounds[1] = lo2;
    }
    __syncthreads();
    const int start = s_bounds[0], end = s_bounds[1];
    const int count = end - start;
    if (count <= 0) return;                              // uniform across block
    const int numTiles = (count + 15) >> 4;

    const int lane = threadIdx.x & 31;
    const int wave = threadIdx.x >> 5;
    const int hb   = wave * 32;
    const int lm   = lane & 15;
    const int hi16 = lane >> 4;
    const int koff = hi16 << 1;

    // Preload B frags of scaled[E,H]: B[k=e][n=h], K=32 -> 8 steps of 4
    v2f bf0[8], bf1[8];
    #pragma unroll
    for (int kk = 0; kk < 8; ++kk) {
        const int e0 = kk * 4 + koff;
        const float* s0 = scaled + e0 * H + hb;
        const float* s1 = s0 + H;
        bf0[kk] = { s0[lm],      s1[lm]      };
        bf1[kk] = { s0[lm + 16], s1[lm + 16] };
    }
    const float wgt0 = weight[hb + lm], wgt1 = weight[hb + 16 + lm];
    const float bi0  = bias[hb + lm],   bi1  = bias[hb + 16 + lm];

    // ---- Sweep A: accumulate sum of squared deviations over this graph
    float acc0 = 0.0f, acc1 = 0.0f;
    for (int t = 0; t < numTiles; ++t) {
        const int ntb   = start + t * 16;
        const int nodeA = ntb + lm;                      // A-frag node (row m)
        const bool va   = nodeA < end;
        v2f af[8];
        #pragma unroll
        for (int kk = 0; kk < 8; ++kk) {
            const int e0 = kk * 4 + koff;
            float u0 = va ? ev[(size_t)nodeA * E + e0]     : 0.0f;
            float u1 = va ? ev[(size_t)nodeA * E + e0 + 1] : 0.0f;
            af[kk] = { u0, u1 };
        }
        v8f c0{}, c1{};
        #pragma unroll
        for (int kk = 0; kk < 8; ++kk) {
            c0 = WMMA_F32(af[kk], bf0[kk], c0);
            c1 = WMMA_F32(af[kk], bf1[kk], c1);
        }
        #pragma unroll
        for (int d = 0; d < 8; ++d) {
            const int node = ntb + d + (hi16 << 3);      // C-layout node
            const bool v   = node < end;
            float x0 = v ? x[(size_t)node * H + hb + lm]      : 0.0f;
            float x1 = v ? x[(size_t)node * H + hb + 16 + lm] : 0.0f;
            float o0 = x0 - c0[d];                       // padded rows: c==0, x==0
            float o1 = x1 - c1[d];
            acc0 += o0 * o0;
            acc1 += o1 * o1;
        }
    }
    // lanes l and l+16 hold the two node-halves of the same h column
    acc0 += __shfl_xor(acc0, 16, 32);
    acc1 += __shfl_xor(acc1, 16, 32);
    const float invc  = 1.0f / (float)count;
    const float rstd0 = rsqrtf(acc0 * invc + EPS);
    const float rstd1 = rsqrtf(acc1 * invc + EPS);

    // ---- Sweep B: recompute mean (x hot in L2), write normalized output
    for (int t = 0; t < numTiles; ++t) {
        const int ntb   = start + t * 16;
        const int nodeA = ntb + lm;
        const bool va   = nodeA < end;
        v2f af[8];
        #pragma unroll
        for (int kk = 0; kk < 8; ++kk) {
            const int e0 = kk * 4 + koff;
            float u0 = va ? ev[(size_t)nodeA * E + e0]     : 0.0f;
            float u1 = va ? ev[(size_t)nodeA * E + e0 + 1] : 0.0f;
            af[kk] = { u0, u1 };
        }
        v8f c0{}, c1{};
        #pragma unroll
        for (int kk = 0; kk < 8; ++kk) {
            c0 = WMMA_F32(af[kk], bf0[kk], c0);
            c1 = WMMA_F32(af[kk], bf1[kk], c1);
        }
        #pragma unroll
        for (int d = 0; d < 8; ++d) {
            const int node = ntb + d + (hi16 << 3);
            if (node < end) {
                const size_t base = (size_t)node * H + hb + lm;
                float o0 = x[base]      - c0[d];
                float o1 = x[base + 16] - c1[d];
                out[base]      = wgt0 * o0 * rstd0 + bi0;
                out[base + 16] = wgt1 * o1 * rstd1 + bi1;
            }
        }
    }
}

// ---------------------------------------------------------------------------
extern "C" void kernel_launch(void* const* d_in, const int* in_sizes, int n_in,
                              void* d_out, int out_size, void* d_ws, size_t ws_size,
                              hipStream_t stream) {
    const float* x         = (const float*)d_in[0];
    const float* evectors  = (const float*)d_in[1];
    const int*   batch     = (const int*)d_in[2];
    const float* weight    = (const float*)d_in[3];
    const float* bias      = (const float*)d_in[4];
    const float* ev_scales = (const float*)d_in[5];
    float* out = (float*)d_out;

    float* contrib = (float*)d_ws;            // E*H floats
    float* scaled  = contrib + E * H;         // E*H floats

    zero_kernel<<<(E * H + 255) / 256, 256, 0, stream>>>(contrib, E * H);
    contribute_kernel<<<K2_BLOCKS, 256, 0, stream>>>(x, evectors, contrib);
    scale_kernel<<<(E * H + 255) / 256, 256, 0, stream>>>(contrib, ev_scales, scaled, E * H);
    norm_kernel<<<G, 256, 0, stream>>>(x, evectors, batch, weight, bias, scaled, out);
}